// Attention_32822140076224
// MI455X (gfx1250) — compile-verified
//
#include <hip/hip_runtime.h>

typedef float v2f __attribute__((ext_vector_type(2)));
typedef float v8f __attribute__((ext_vector_type(8)));
typedef int   v4i __attribute__((ext_vector_type(4)));

constexpr int kB  = 2, kH = 8, kS = 2048, kD = 64;
constexpr int kBH = kB * kH;        // 16 (b,h) pairs
constexpr int kQT = kS / 16;        // 128 16-row tiles per head
constexpr float kScale = 0.125f;    // 1/sqrt(64)

// ---- async global -> LDS staging (gfx1250 ASYNCcnt path, with fallback) ----
__device__ __forceinline__ void stage_b128(const float* g, float* l)
{
#if defined(__gfx1250__) && __has_builtin(__builtin_amdgcn_global_load_async_to_lds_b128)
    __builtin_amdgcn_global_load_async_to_lds_b128(
        (__attribute__((address_space(1))) v4i*)g,
        (__attribute__((address_space(3))) v4i*)l, 0, 0);
#else
    *(float4*)l = *(const float4*)g;   // cooperative sync copy fallback
#endif
}

__device__ __forceinline__ void wait_stage()
{
#if defined(__gfx1250__) && __has_builtin(__builtin_amdgcn_global_load_async_to_lds_b128)
#if __has_builtin(__builtin_amdgcn_s_wait_asynccnt)
    __builtin_amdgcn_s_wait_asynccnt(0);
#else
    asm volatile("s_wait_asynccnt 0" ::: "memory");
#endif
#endif
}

// Fused: a = mask ? -1e9 : QK^T/8  (written straight into attn_p region)
//        attn_v = a @ V            (faithful to the reference's pre-softmax bug)
// 4 waves/block share one head; K/V 16x64 tiles staged once per block in LDS,
// double-buffered so the async copy of tile kt+1 overlaps the WMMAs of tile kt.
__global__ __launch_bounds__(128)
void attn_qk_av_kernel(const float* __restrict__ Q,
                       const float* __restrict__ K,
                       const float* __restrict__ V,
                       const unsigned char* __restrict__ M,
                       float* __restrict__ out_v,
                       float* __restrict__ out_p)
{
    __shared__ float kbuf[2][16 * 64];          // 2 x 4KB K tiles
    __shared__ float vbuf[2][16 * 64];          // 2 x 4KB V tiles
    __shared__ float xpose[4][256];             // per-wave 16x16 transpose tile

    const int wave = threadIdx.x >> 5;
    const int lane = threadIdx.x & 31;
    const int gw   = blockIdx.x * 4 + wave;     // global wave id
    const int bh   = gw >> 7;                   // head index 0..15 (block-uniform)
    const int q16  = gw & (kQT - 1);            // q tile 0..127
    const int n    = lane & 15;                 // row (A) / column (B,C,D) index
    const int hi   = lane >> 4;                 // half-wave select
    float* tile = xpose[wave];

    const float* Qh = Q + (size_t)bh * kS * kD;
    const float* Kh = K + (size_t)bh * kS * kD;
    const float* Vh = V + (size_t)bh * kS * kD;
    const unsigned char* Mh = M + (size_t)bh * kS * kS;
    float* Ph = out_p + (size_t)bh * kS * kS;
    float* Oh = out_v + (size_t)bh * kS * kD;

    // Q tile as 16 A-operands (16x4 f32 layout), resident for the whole sweep:
    // lane (n, hi): A0 = Q[n][4s+2hi], A1 = Q[n][4s+2hi+1]  -> contiguous float2
    v2f aq[16];
    const float* qrow = Qh + (size_t)(q16 * 16 + n) * kD + 2 * hi;
    #pragma unroll
    for (int s = 0; s < 16; ++s) aq[s] = *(const v2f*)(qrow + 4 * s);

    // Prologue: stage tile 0 (each of 128 threads moves 2x16B of K and of V).
    const int tf = threadIdx.x * 4;             // flat float index, 0..508
    stage_b128(Kh + tf,       &kbuf[0][tf]);
    stage_b128(Kh + tf + 512, &kbuf[0][tf + 512]);
    stage_b128(Vh + tf,       &vbuf[0][tf]);
    stage_b128(Vh + tf + 512, &vbuf[0][tf + 512]);
    wait_stage();
    __syncthreads();

    const v8f vzero = {0.f, 0.f, 0.f, 0.f, 0.f, 0.f, 0.f, 0.f};
    v8f vacc[4] = {vzero, vzero, vzero, vzero};  // 16(q) x 64(d) attn_v accum

    for (int kt = 0; kt < kQT; ++kt) {
        const int cur = kt & 1;

        // Kick off async staging of the next K/V tiles into the other buffer.
        if (kt + 1 < kQT) {
            const float* gk = Kh + (size_t)(kt + 1) * 1024;
            const float* gv = Vh + (size_t)(kt + 1) * 1024;
            stage_b128(gk + tf,       &kbuf[1 - cur][tf]);
            stage_b128(gk + tf + 512, &kbuf[1 - cur][tf + 512]);
            stage_b128(gv + tf,       &vbuf[1 - cur][tf]);
            stage_b128(gv + tf + 512, &vbuf[1 - cur][tf + 512]);
        }

        // ---- score tile: D(16x16) += A(Q 16x4) x B(K^T 4x16), 16 steps ----
        // B operand from LDS: lane (n,hi) reads K[n][4s+2hi..+1] (ds_load_b64)
        const float* kb = &kbuf[cur][n * 64 + 2 * hi];
        v2f bk[16];
        #pragma unroll
        for (int s = 0; s < 16; ++s) bk[s] = *(const v2f*)(kb + 4 * s);

        v8f sacc = vzero;
        #pragma unroll
        for (int s = 0; s < 16; ++s)
            sacc = __builtin_amdgcn_wmma_f32_16x16x4_f32(
                false, aq[s], false, bk[s], (short)0, sacc, false, false);

        // ---- scale + mask + store pre-softmax scores (this IS attn_p input) ----
        #pragma unroll
        for (int r = 0; r < 8; ++r) {
            const int qq = q16 * 16 + r + 8 * hi;          // C/D row = r + 8*hi
            const size_t off = (size_t)qq * kS + kt * 16 + n;
            const float v = Mh[off] ? -1e9f : sacc[r] * kScale;
            sacc[r] = v;
            Ph[off] = v;
        }

        // ---- relayout D-layout -> A-layout through LDS (cross-lane transpose) ----
        #pragma unroll
        for (int r = 0; r < 8; ++r) tile[(r + 8 * hi) * 16 + n] = sacc[r];
        asm volatile("s_wait_dscnt 0" ::: "memory");        // wave-internal DS RAW
        v2f a_op[4];
        #pragma unroll
        for (int s2 = 0; s2 < 4; ++s2)
            a_op[s2] = *(const v2f*)&tile[n * 16 + 4 * s2 + 2 * hi];

        // ---- attn_v accumulate: 4 d-chunks x 4 k-steps of wmma ----
        // B operand rows sit 64 dwords apart in LDS -> ds_load_2addr_b32 pairs
        #pragma unroll
        for (int dc = 0; dc < 4; ++dc) {
            #pragma unroll
            for (int s2 = 0; s2 < 4; ++s2) {
                const float* vp = &vbuf[cur][(4 * s2 + 2 * hi) * 64 + dc * 16 + n];
                v2f bv;
                bv.x = vp[0];
                bv.y = vp[64];
                vacc[dc] = __builtin_amdgcn_wmma_f32_16x16x4_f32(
                    false, a_op[s2], false, bv, (short)0, vacc[dc], false, false);
            }
        }

        wait_stage();        // this wave's async copies for kt+1 are done
        __syncthreads();     // whole block done reading cur / writing next
    }

    // ---- write attn_v tile (16 x 64) ----
    #pragma unroll
    for (int dc = 0; dc < 4; ++dc) {
        #pragma unroll
        for (int r = 0; r < 8; ++r)
            Oh[(size_t)(q16 * 16 + r + 8 * hi) * kD + dc * 16 + n] = vacc[dc][r];
    }
}

// In-place row softmax over the attn_p region: one 256-thread block per row,
// row (2048 f32) held in registers -> exactly one read + one write of 268 MB.
__global__ __launch_bounds__(256)
void softmax_rows_kernel(float* __restrict__ P)
{
    __shared__ float red[8];
    float* prow = P + (size_t)blockIdx.x * kS;
    const int t = threadIdx.x;

    float x[8];
    #pragma unroll
    for (int i = 0; i < 8; ++i) x[i] = prow[t + i * 256];

    float mx = -3.4e38f;
    #pragma unroll
    for (int i = 0; i < 8; ++i) mx = fmaxf(mx, x[i]);
    #pragma unroll
    for (int off = 16; off > 0; off >>= 1)
        mx = fmaxf(mx, __shfl_xor(mx, off, 32));
    if ((t & 31) == 0) red[t >> 5] = mx;
    __syncthreads();
    #pragma unroll
    for (int i = 0; i < 8; ++i) mx = fmaxf(mx, red[i]);

    float s = 0.f;
    #pragma unroll
    for (int i = 0; i < 8; ++i) { x[i] = __expf(x[i] - mx); s += x[i]; }
    #pragma unroll
    for (int off = 16; off > 0; off >>= 1) s += __shfl_xor(s, off, 32);
    __syncthreads();                         // red[] reuse
    if ((t & 31) == 0) red[t >> 5] = s;
    __syncthreads();
    s = 0.f;
    #pragma unroll
    for (int i = 0; i < 8; ++i) s += red[i];
    const float inv = 1.f / s;
    #pragma unroll
    for (int i = 0; i < 8; ++i) prow[t + i * 256] = x[i] * inv;
}

extern "C" void kernel_launch(void* const* d_in, const int* in_sizes, int n_in,
                              void* d_out, int out_size, void* d_ws, size_t ws_size,
                              hipStream_t stream)
{
    const float* Q = (const float*)d_in[0];
    const float* K = (const float*)d_in[1];
    const float* V = (const float*)d_in[2];
    const unsigned char* M = (const unsigned char*)d_in[3];  // jnp.bool_ -> 1 byte
    float* out_v = (float*)d_out;                            // (B,H,S,D) first
    float* out_p = out_v + (size_t)kBH * kS * kD;            // then (B,H,S,S)

    // 16 heads * 128 q-tiles = 2048 waves; 4 waves per 128-thread block
    attn_qk_av_kernel<<<(kBH * kQT) / 4, 128, 0, stream>>>(Q, K, V, M, out_v, out_p);
    // one block per (b,h,q) row
    softmax_rows_kernel<<<kBH * kS, 256, 0, stream>>>(out_p);
}